// Transformer_6184752906878
// MI455X (gfx1250) — compile-verified
//
#include <hip/hip_runtime.h>

// out[b,s,:] = W[x[b,s],:] + PE[s,:]
// x: [8,2048] int32, W: [50257,1024] f32, out: [8,2048,1024] f32
// Memory-bound: ~128 MiB moved -> ~5.5us floor at 23.3 TB/s.

#define SEQ   2048
#define EMB   1024
#define BATCH 8
#define ROWS  (BATCH * SEQ)

typedef float v4f __attribute__((ext_vector_type(4)));
typedef int   v4i __attribute__((ext_vector_type(4)));

typedef __attribute__((address_space(1))) v4i* gv4i_p;  // global
typedef __attribute__((address_space(3))) v4i* lv4i_p;  // LDS

#if __has_builtin(__builtin_amdgcn_global_load_async_to_lds_b128)
#define USE_ASYNC_LDS 1
#endif

// PE(s, d): even d -> sin(s / 10000^(d/EMB)), odd d -> cos of same angle.
__device__ __forceinline__ v4f pe_quad(int s, int t) {
    const float K = -2.0f * 13.287712379549449f / (float)EMB; // -2*log2(10000)/EMB
    float i0 = (float)(2 * t);
    float i1 = i0 + 1.0f;
    float a0 = (float)s * exp2f(K * i0);   // pos / 10000^(2*i0/EMB)
    float a1 = (float)s * exp2f(K * i1);
    float s0, c0, s1, c1;
    __sincosf(a0, &s0, &c0);
    __sincosf(a1, &s1, &c1);
    v4f r;
    r.x = s0; r.y = c0; r.z = s1; r.w = c1;
    return r;
}

// Kernel 1: build PE table [SEQ][EMB] into workspace (8 MiB, L2-resident).
__global__ void __launch_bounds__(256) pe_init_kernel(float* __restrict__ pe) {
    const int s = blockIdx.x;    // position
    const int t = threadIdx.x;   // float4 index within row (256*4 = 1024)
    ((v4f*)(pe + (size_t)s * EMB))[t] = pe_quad(s, t);
}

// Kernel 2: gather embedding row + add PE row. One block per (b,s) row.
__global__ void __launch_bounds__(256) gather_add_kernel(
        const int*   __restrict__ x,
        const float* __restrict__ W,
        const float* __restrict__ pe,
        float*       __restrict__ out) {
    const int row = blockIdx.x;            // b*SEQ + s
    const int t   = threadIdx.x;
    const int s   = row & (SEQ - 1);
    // token id is uniform across the block -> scalar load path
    const int tok = __builtin_amdgcn_readfirstlane(x[row]);

#if defined(USE_ASYNC_LDS)
    // Async-copy the PE row into LDS (ASYNCcnt path) while the W gather load
    // is in flight. Each lane copies and later reads its own 16B chunk, so a
    // per-wave s_wait_asynccnt suffices (no workgroup barrier needed).
    __shared__ float lds_pe[EMB];
    {
        float* g = const_cast<float*>(pe) + (size_t)s * EMB + 4 * t;
        float* l = lds_pe + 4 * t;
        __builtin_amdgcn_global_load_async_to_lds_b128(
            (gv4i_p)(void*)g,
            (lv4i_p)(void*)l,
            /*offset=*/0, /*cpol=*/0);
    }
#endif

    // 128-bit coalesced gather of the embedding row (L2-cacheable: token
    // repeats within the batch hit in the 192MB L2).
    v4f w = ((const v4f*)(W + (size_t)tok * EMB))[t];

#if defined(USE_ASYNC_LDS)
    asm volatile("s_wait_asynccnt 0" ::: "memory");
    v4f p = ((const v4f*)lds_pe)[t];
#else
    v4f p = ((const v4f*)(pe + (size_t)s * EMB))[t];
#endif

    v4f o = w + p;
    // Non-temporal 128-bit store: the 64 MiB output stream is write-once,
    // keep it from evicting the L2-resident PE table / hot W rows.
    __builtin_nontemporal_store(o, (v4f*)(out + (size_t)row * EMB) + t);
}

// Fallback if workspace is too small: recompute PE per row on the fly.
__global__ void __launch_bounds__(256) gather_add_fused_kernel(
        const int*   __restrict__ x,
        const float* __restrict__ W,
        float*       __restrict__ out) {
    const int row = blockIdx.x;
    const int t   = threadIdx.x;
    const int s   = row & (SEQ - 1);
    const int tok = __builtin_amdgcn_readfirstlane(x[row]);

    v4f w = ((const v4f*)(W + (size_t)tok * EMB))[t];
    v4f p = pe_quad(s, t);
    v4f o = w + p;
    __builtin_nontemporal_store(o, (v4f*)(out + (size_t)row * EMB) + t);
}

extern "C" void kernel_launch(void* const* d_in, const int* in_sizes, int n_in,
                              void* d_out, int out_size, void* d_ws, size_t ws_size,
                              hipStream_t stream) {
    const int*   x   = (const int*)d_in[0];    // [8,2048] int32 token ids
    const float* W   = (const float*)d_in[1];  // [50257,1024] f32
    float*       out = (float*)d_out;          // [8,2048,1024] f32

    const size_t pe_bytes = (size_t)SEQ * EMB * sizeof(float); // 8 MiB

    if (ws_size >= pe_bytes) {
        float* pe = (float*)d_ws;
        pe_init_kernel<<<SEQ, 256, 0, stream>>>(pe);
        gather_add_kernel<<<ROWS, 256, 0, stream>>>(x, W, pe, out);
    } else {
        gather_add_fused_kernel<<<ROWS, 256, 0, stream>>>(x, W, out);
    }
}